// NeuralRadiance_8211977470216
// MI455X (gfx1250) — compile-verified
//
#include <hip/hip_runtime.h>
#include <stdint.h>

typedef __attribute__((ext_vector_type(2))) float v2f;
typedef __attribute__((ext_vector_type(4))) float v4f;
typedef __attribute__((ext_vector_type(8))) float v8f;

#define NWAVES 4
#define BLOCK  (NWAVES * 32)
#define HS     68    // activation-tile row stride (floats): (4*ln+c)%64 -> conflict-free b64 col reads
#define W1S    128   // k-pair row stride (dwords) for swizzled W1 (10 rows)
#define W2S    160   // k-pair row stride for swizzled W2 (32 rows): ==32 mod 64 -> conflict-free
#define W3S    32    // k-pair row stride for swizzled W3 (32 rows): ==32 mod 64 -> conflict-free

__global__ __launch_bounds__(BLOCK) void nerf_mlp_wmma_f32(
    const float* __restrict__ pos, const float* __restrict__ nrm,
    const float* __restrict__ emb,
    const float* __restrict__ W1, const float* __restrict__ b1,
    const float* __restrict__ W2, const float* __restrict__ b2,
    const float* __restrict__ W3, const float* __restrict__ b3,
    float* __restrict__ out, int npts, int ngroups)
{
    // Swizzled weight layout: element (k, n) at  [ (k>>1)*stride + 2*n + (k&1) ]
    // => one lane's B fragment (rows k,k+1 at col n) is a single aligned 8B ds_load_b64.
    __shared__ float sW1s[10 * W1S];          // 20x64 padded (row 19 = 0), swizzled
    __shared__ float sW2s[32 * W2S];          // 64x64 swizzled
    __shared__ float sW3s[32 * W3S];          // 64x16 padded (cols 3..15 = 0), swizzled
    __shared__ float sB1[64];
    __shared__ float sB2[64];
    __shared__ float sB3[16];
    __shared__ float bufA[NWAVES][16 * HS];   // x tile, later h2 tile
    __shared__ float bufB[NWAVES][16 * HS];   // h1 tile

    const int tid = threadIdx.x;

    // ---- stage + swizzle weights into LDS (once per block, amortized by persistent loop) ----
    for (int i = tid; i < 20 * 64; i += BLOCK) {
        const int k = i >> 6, n = i & 63;
        sW1s[(k >> 1) * W1S + 2 * n + (k & 1)] = (k < 19) ? W1[i] : 0.0f;
    }
    for (int i = tid; i < 64 * 64; i += BLOCK) {
        const int k = i >> 6, n = i & 63;
        sW2s[(k >> 1) * W2S + 2 * n + (k & 1)] = W2[i];
    }
    for (int i = tid; i < 64 * 16; i += BLOCK) {
        const int k = i >> 4, c = i & 15;
        sW3s[(k >> 1) * W3S + 2 * c + (k & 1)] = (c < 3) ? W3[k * 3 + c] : 0.0f;
    }
    if (tid < 64) { sB1[tid] = b1[tid]; sB2[tid] = b2[tid]; }
    if (tid < 16) sB3[tid] = (tid < 3) ? b3[tid] : 0.0f;
    __syncthreads();

    const int wave = tid >> 5;
    const int lane = tid & 31;
    const int half = lane >> 4;     // 0: lanes 0-15, 1: lanes 16-31
    const int ln   = lane & 15;

    float* xt = bufA[wave];
    float* ht = bufB[wave];

    // hoist loop-invariant biases into registers
    float bias1[4], bias2[4];
    #pragma unroll
    for (int nt = 0; nt < 4; ++nt) { bias1[nt] = sB1[nt * 16 + ln]; bias2[nt] = sB2[nt * 16 + ln]; }
    const float bias3 = sB3[ln];

    // ---- persistent loop over tile groups ----
    for (int g = blockIdx.x; g < ngroups; g += gridDim.x) {
        const int  tile   = g * NWAVES + wave;   // 16 points per wave
        const int  base   = tile * 16;
        const bool active = (base + 16) <= npts;

        // ---- feature gather: lanes 0..15 each build one row of x[16 x 20] ----
        if (active && lane < 16) {
            const int p = base + ln;
            const float px = pos[3 * p + 0];
            const float py = pos[3 * p + 1];
            const float pz = pos[3 * p + 2];
            // trunc-toward-zero cast == astype(int32); uint mul == int32 wraparound
            const uint32_t sx = (uint32_t)(int)(px * 8.0f);
            const uint32_t sy = (uint32_t)(int)(py * 8.0f);
            const uint32_t sz = (uint32_t)(int)(pz * 8.0f);
            const uint32_t h  = (sx * 73856093u) ^ (sy * 19349663u) ^ (sz * 83492791u);
            const uint32_t idx = h & 32767u;     // TABLE-1

            const v4f* e  = (const v4f*)(emb + (size_t)idx * 16);
            v4f e0 = e[0], e1 = e[1], e2 = e[2], e3 = e[3];
            v4f* xr = (v4f*)(xt + ln * HS);      // ln*272 B : 16B aligned
            xr[0] = e0; xr[1] = e1; xr[2] = e2; xr[3] = e3;
            xt[ln * HS + 16] = nrm[3 * p + 0];
            xt[ln * HS + 17] = nrm[3 * p + 1];
            xt[ln * HS + 18] = nrm[3 * p + 2];
            xt[ln * HS + 19] = 0.0f;
        }
        __syncthreads();

        const v2f* xrow = (const v2f*)(xt + ln * HS);   // A-frag source, 8B aligned (HS even)
        const v2f* hrow = (const v2f*)(ht + ln * HS);

        // ---- layer 1: x[16x20] @ W1[20x64] + b1, relu -> ht ----
        #pragma unroll
        for (int nt = 0; nt < 4; ++nt) {
            const float bb = bias1[nt];
            v8f acc = { bb, bb, bb, bb, bb, bb, bb, bb };
            #pragma unroll
            for (int kt = 0; kt < 5; ++kt) {
                const int kp = kt * 2 + half;                 // k-pair index
                const v2f a = xrow[kp];
                const v2f b = *(const v2f*)(sW1s + kp * W1S + 2 * (nt * 16 + ln));
                acc = __builtin_amdgcn_wmma_f32_16x16x4_f32(
                    false, a, false, b, (short)0, acc, false, false);
            }
            #pragma unroll
            for (int r = 0; r < 8; ++r) {
                const int m = half * 8 + r;                   // D layout: M = r + 8*half
                ht[m * HS + nt * 16 + ln] = fmaxf(acc[r], 0.0f);   // single v_max_num_f32
            }
        }
        __syncthreads();

        // ---- layer 2: h1[16x64] @ W2[64x64] + b2, relu -> xt (x consumed) ----
        #pragma unroll
        for (int nt = 0; nt < 4; ++nt) {
            const float bb = bias2[nt];
            v8f acc = { bb, bb, bb, bb, bb, bb, bb, bb };
            #pragma unroll
            for (int kt = 0; kt < 16; ++kt) {
                const int kp = kt * 2 + half;
                const v2f a = hrow[kp];
                const v2f b = *(const v2f*)(sW2s + kp * W2S + 2 * (nt * 16 + ln));
                acc = __builtin_amdgcn_wmma_f32_16x16x4_f32(
                    false, a, false, b, (short)0, acc, false, false);
            }
            #pragma unroll
            for (int r = 0; r < 8; ++r) {
                const int m = half * 8 + r;
                xt[m * HS + nt * 16 + ln] = fmaxf(acc[r], 0.0f);
            }
        }
        __syncthreads();

        // ---- layer 3: h2[16x64] @ W3pad[64x16] + b3, sigmoid -> out (n<3) ----
        {
            v8f acc = { bias3, bias3, bias3, bias3, bias3, bias3, bias3, bias3 };
            #pragma unroll
            for (int kt = 0; kt < 16; ++kt) {
                const int kp = kt * 2 + half;
                const v2f a = xrow[kp];
                const v2f b = *(const v2f*)(sW3s + kp * W3S + 2 * ln);
                acc = __builtin_amdgcn_wmma_f32_16x16x4_f32(
                    false, a, false, b, (short)0, acc, false, false);
            }
            if (active && ln < 3) {
                #pragma unroll
                for (int r = 0; r < 8; ++r) {
                    const int m = half * 8 + r;
                    const float v = acc[r];
                    out[(size_t)(base + m) * 3 + ln] = 1.0f / (1.0f + __expf(-v));
                }
            }
        }
        __syncthreads();   // protect xt/ht against next iteration's writes
    }
}

extern "C" void kernel_launch(void* const* d_in, const int* in_sizes, int n_in,
                              void* d_out, int out_size, void* d_ws, size_t ws_size,
                              hipStream_t stream) {
    const float* pos = (const float*)d_in[0];
    const float* nrm = (const float*)d_in[1];
    const float* emb = (const float*)d_in[2];
    const float* W1  = (const float*)d_in[3];
    const float* b1  = (const float*)d_in[4];
    const float* W2  = (const float*)d_in[5];
    const float* b2  = (const float*)d_in[6];
    const float* W3  = (const float*)d_in[7];
    const float* b3  = (const float*)d_in[8];
    float* out = (float*)d_out;

    const int npts    = in_sizes[0] / 3;
    const int ntiles  = (npts + 15) / 16;
    const int ngroups = (ntiles + NWAVES - 1) / NWAVES;
    const int nblocks = ngroups < 2048 ? ngroups : 2048;   // persistent blocks

    hipLaunchKernelGGL(nerf_mlp_wmma_f32, dim3(nblocks), dim3(BLOCK), 0, stream,
                       pos, nrm, emb, W1, b1, W2, b2, W3, b3, out, npts, ngroups);
}